// GCN_61108794688064
// MI455X (gfx1250) — compile-verified
//
#include <hip/hip_runtime.h>
#include <hip/hip_bf16.h>
#include <math.h>

typedef __attribute__((ext_vector_type(2))) float v2f;
typedef __attribute__((ext_vector_type(8))) float v8f;

#define F_IN 500
#define HID  16
#define NCLS 10

// ---------------------------------------------------------------------------
// Degree / normalization helpers
// ---------------------------------------------------------------------------
__global__ void k_fill(float* __restrict__ p, float v, int n) {
  int i = blockIdx.x * blockDim.x + threadIdx.x;
  if (i < n) p[i] = v;
}

__global__ void k_count_deg(const long long* __restrict__ dst,
                            float* __restrict__ deg, long long E) {
  long long e = (long long)blockIdx.x * blockDim.x + threadIdx.x;
  if (e < E) atomicAdd(&deg[dst[e]], 1.0f);
}

__global__ void k_rsqrt(float* __restrict__ p, int n) {
  int i = blockIdx.x * blockDim.x + threadIdx.x;
  if (i < n) {
    float d = p[i];
    p[i] = (d > 0.f) ? rsqrtf(d) : 0.f;
  }
}

// ---------------------------------------------------------------------------
// Skinny GEMM via V_WMMA_F32_16X16X4_F32.
// One wave computes a 16-row tile of D = A[N x K] * B[K x BCOLS] (B zero-
// padded to 16 columns in LDS). Output row stride fixed at 16 (HID).
// A-layout: lane L -> row L%16, K-pair 2*(L/16)   (one global_load_b64 / step)
// B-layout: lane L -> col L%16, K-pair 2*(L/16)   (one ds_load_b64 / step)
// C/D     : VGPR r -> row r + 8*(L>=16), col L%16
// B is staged transposed+padded in LDS so the inner loop carries no
// predication (EXEC stays all-1s, as WMMA requires) and no redundant loads.
// Store epilogue uses a wave-uniform full-tile fast path so the common case
// has no per-store EXEC manipulation.
// ---------------------------------------------------------------------------
template <int K, int LDA, int BCOLS, int LDB>
__global__ void k_gemm_wmma(const float* __restrict__ A, const float* __restrict__ B,
                            float* __restrict__ out, int N) {
  constexpr int KS = K + 2;               // pad row stride: bank-conflict-free
  __shared__ float BT[16 * KS];

  // Stage BT[n][k] = (n < BCOLS) ? B[k][n] : 0   (coalesced over threads)
  for (int i = threadIdx.x; i < 16 * K; i += blockDim.x) {
    int n = i & 15, k = i >> 4;
    BT[n * KS + k] = (n < BCOLS) ? B[(size_t)k * LDB + n] : 0.f;
  }
  __syncthreads();

  int wave   = (blockIdx.x * blockDim.x + threadIdx.x) >> 5;
  int lane   = threadIdx.x & 31;
  int ntiles = (N + 15) >> 4;
  if (wave >= ntiles) return;             // wave-uniform: EXEC stays all-1s

  int hi   = lane >> 4;                   // 0 or 1 (half-wave)
  int m    = lane & 15;                   // row (A) / col (B) within tile
  int base = wave * 16;
  int row  = base + m;
  int rowc = (row < N) ? row : (N - 1);   // clamp instead of predicating
  const float* arow = A + (size_t)rowc * LDA;
  const float* brow = &BT[m * KS];

  v8f acc = {};
#pragma unroll 5
  for (int k = 0; k < K; k += 4) {
    int kk = k + 2 * hi;
    v2f a = *(const v2f*)(arow + kk);     // 8B-aligned: LDA even, kk even
    v2f b = *(const v2f*)(brow + kk);     // LDS, 8B-aligned
    acc = __builtin_amdgcn_wmma_f32_16x16x4_f32(
        /*neg_a=*/false, a, /*neg_b=*/false, b,
        /*c_mod=*/(short)0, acc, /*reuse_a=*/false, /*reuse_b=*/false);
  }

  // Epilogue: lane writes rows base+8*hi+r, column m.
  float* op = out + (size_t)(base + 8 * hi) * HID + m;
  if (base + 16 <= N) {                   // wave-uniform full-tile fast path
#pragma unroll
    for (int r = 0; r < 8; ++r) op[r * HID] = acc[r];
  } else {                                // tail tile (not hit when N%16==0)
#pragma unroll
    for (int r = 0; r < 8; ++r)
      if (base + 8 * hi + r < N) op[r * HID] = acc[r];
  }
}

// ---------------------------------------------------------------------------
// Self-loop initialization: out = h * dinv[node]^2  (the self-edge term)
// ---------------------------------------------------------------------------
__global__ void k_selfloop16(const float* __restrict__ h, const float* __restrict__ dinv,
                             float* __restrict__ out, int N) {
  int i = blockIdx.x * blockDim.x + threadIdx.x;
  if (i >= N * HID) return;
  float w = dinv[i >> 4];
  out[i] = h[i] * (w * w);
}

__global__ void k_selfloop10(const float* __restrict__ h2, const float* __restrict__ dinv,
                             float* __restrict__ out, int N) {
  int i = blockIdx.x * blockDim.x + threadIdx.x;
  if (i >= N * NCLS) return;
  int node = i / NCLS, c = i % NCLS;
  float w = dinv[node];
  out[i] = h2[(size_t)node * HID + c] * (w * w);
}

// ---------------------------------------------------------------------------
// Edge gather-scale-scatter: out[dst] += h[src] * dinv[src]*dinv[dst]
// Feature maps are L2-resident (<=6.4 MB) so the f32 atomics stay on-chip.
// Gather h[src] with vector loads into registers, then issue the atomics.
// ---------------------------------------------------------------------------
template <int F, int HS, int OS>
__global__ void k_agg(const long long* __restrict__ src, const long long* __restrict__ dst,
                      const float* __restrict__ dinv, const float* __restrict__ h,
                      float* __restrict__ out, long long E) {
  long long e = (long long)blockIdx.x * blockDim.x + threadIdx.x;
  if (e >= E) return;
  long long s = src[e], d = dst[e];
  float w = dinv[s] * dinv[d];
  const float* hp = h + (size_t)s * HS;   // 64B-aligned rows (HS = 16)
  float* op = out + (size_t)d * OS;
  float vbuf[F];
#pragma unroll
  for (int c = 0; c < F; c += 2) {        // F is even: 8B vector gathers
    v2f t = *(const v2f*)(hp + c);
    vbuf[c] = t.x * w;
    vbuf[c + 1] = t.y * w;
  }
#pragma unroll
  for (int c = 0; c < F; ++c) atomicAdd(op + c, vbuf[c]);
}

// ---------------------------------------------------------------------------
// Epilogues
// ---------------------------------------------------------------------------
__global__ void k_bias_relu(float* __restrict__ g, const float* __restrict__ b, int N) {
  int i = blockIdx.x * blockDim.x + threadIdx.x;
  if (i >= N * HID) return;
  float v = g[i] + b[i & (HID - 1)];
  g[i] = (v > 0.f) ? v : 0.f;
}

__global__ void k_bias_logsoftmax(float* __restrict__ out, const float* __restrict__ b, int N) {
  int i = blockIdx.x * blockDim.x + threadIdx.x;
  if (i >= N) return;
  float v[NCLS];
  float mx = -INFINITY;
#pragma unroll
  for (int c = 0; c < NCLS; ++c) {
    v[c] = out[(size_t)i * NCLS + c] + b[c];
    mx = fmaxf(mx, v[c]);
  }
  float s = 0.f;
#pragma unroll
  for (int c = 0; c < NCLS; ++c) s += __expf(v[c] - mx);
  float lse = mx + __logf(s);
#pragma unroll
  for (int c = 0; c < NCLS; ++c) out[(size_t)i * NCLS + c] = v[c] - lse;
}

// ---------------------------------------------------------------------------
// Driver
// ---------------------------------------------------------------------------
extern "C" void kernel_launch(void* const* d_in, const int* in_sizes, int n_in,
                              void* d_out, int out_size, void* d_ws, size_t ws_size,
                              hipStream_t stream) {
  const float*     x   = (const float*)d_in[0];
  const long long* ei  = (const long long*)d_in[1];   // int64 per reference
  const float*     W1  = (const float*)d_in[2];
  const float*     b1  = (const float*)d_in[3];
  const float*     W2  = (const float*)d_in[4];
  const float*     b2  = (const float*)d_in[5];
  float*           outp = (float*)d_out;

  int       N = in_sizes[0] / F_IN;
  long long E = in_sizes[1] / 2;
  const long long* srcp = ei;
  const long long* dstp = ei + E;

  // Workspace layout (floats): dinv[N] | h1[N*16] | g1[N*16] | h2[N*16]
  float* dinv = (float*)d_ws;
  float* h1   = dinv + N;
  float* g1   = h1 + (size_t)N * HID;
  float* h2   = g1 + (size_t)N * HID;

  const int T = 256;
  int gN  = (N + T - 1) / T;
  int gE  = (int)((E + T - 1) / T);
  int gNH = (N * HID + T - 1) / T;
  int gNC = (N * NCLS + T - 1) / T;
  int tiles = (N + 15) / 16;
  int gW  = (tiles * 32 + T - 1) / T;    // one wave32 per 16-row tile

  // Degrees (self-loop => start at 1) and D^{-1/2}
  k_fill<<<gN, T, 0, stream>>>(dinv, 1.0f, N);
  k_count_deg<<<gE, T, 0, stream>>>(dstp, dinv, E);
  k_rsqrt<<<gN, T, 0, stream>>>(dinv, N);

  // Layer 1: h1 = x @ W1 (fp32 WMMA), then normalized aggregation + bias + relu
  k_gemm_wmma<F_IN, F_IN, HID, HID><<<gW, T, 0, stream>>>(x, W1, h1, N);
  k_selfloop16<<<gNH, T, 0, stream>>>(h1, dinv, g1, N);
  k_agg<HID, HID, HID><<<gE, T, 0, stream>>>(srcp, dstp, dinv, h1, g1, E);
  k_bias_relu<<<gNH, T, 0, stream>>>(g1, b1, N);

  // Layer 2: h2 = g1 @ W2 (W2 zero-padded to 16 cols in LDS)
  k_gemm_wmma<HID, HID, NCLS, NCLS><<<gW, T, 0, stream>>>(g1, W2, h2, N);
  k_selfloop10<<<gNC, T, 0, stream>>>(h2, dinv, outp, N);
  k_agg<NCLS, HID, NCLS><<<gE, T, 0, stream>>>(srcp, dstp, dinv, h2, outp, E);

  // Bias + log_softmax in place on d_out
  k_bias_logsoftmax<<<gN, T, 0, stream>>>(outp, b2, N);
}